// Comparator4Bit_76312978916078
// MI455X (gfx1250) — compile-verified
//
#include <hip/hip_runtime.h>
#include <stdint.h>

#define TPB 256

// --- gfx1250 async global->LDS copy (ASYNCcnt-tracked), per-lane 16B ---
__device__ __forceinline__ void async_ld_b128(uint32_t lds_byte_addr, const void* gptr) {
  asm volatile("global_load_async_to_lds_b128 %0, %1, off"
               :
               : "v"(lds_byte_addr),
                 "v"((unsigned long long)(uintptr_t)gptr)
               : "memory");
}

__device__ __forceinline__ void wait_async_le2() {
  asm volatile("s_wait_asynccnt 0x2" ::: "memory");
}
__device__ __forceinline__ void wait_async_le0() {
  asm volatile("s_wait_asynccnt 0x0" ::: "memory");
}

__global__ __launch_bounds__(TPB, 1) void cmp4bit_async_kernel(
    const float4* __restrict__ A, const float4* __restrict__ B,
    float* __restrict__ out_gt, float* __restrict__ out_eq,
    int n_rows, int iters, int stride_rows)
{
  __shared__ float4 sA[2][TPB];
  __shared__ float4 sB[2][TPB];

  const int tid = threadIdx.x;
  int row = blockIdx.x * TPB + tid;

  const uint32_t ldsA0 = (uint32_t)(uintptr_t)&sA[0][tid];
  const uint32_t ldsA1 = (uint32_t)(uintptr_t)&sA[1][tid];
  const uint32_t ldsB0 = (uint32_t)(uintptr_t)&sB[0][tid];
  const uint32_t ldsB1 = (uint32_t)(uintptr_t)&sB[1][tid];

  // Prefetch tile 0 (clamped address; out-of-range rows do redundant work)
  {
    const int r = (row < n_rows) ? row : (n_rows - 1);
    async_ld_b128(ldsA0, A + r);
    async_ld_b128(ldsB0, B + r);
  }

  int next = row + stride_rows;
  for (int i = 0; i < iters; ++i) {
    const int buf = i & 1;

    if (i + 1 < iters) {
      // Prefetch next tile into the other buffer, then wait for the
      // current tile's 2 loads (in-order completion => ASYNCcnt <= 2).
      const int r = (next < n_rows) ? next : (n_rows - 1);
      async_ld_b128(buf ? ldsA0 : ldsA1, A + r);
      async_ld_b128(buf ? ldsB0 : ldsB1, B + r);
      wait_async_le2();
    } else {
      wait_async_le0();
    }

    const float4 a = buf ? sA[1][tid] : sA[0][tid];
    const float4 b = buf ? sB[1][tid] : sB[0][tid];

    // Element 0 is the MSB in the reference. Bits are exact 0/1 floats,
    // so the weighted sum is exact and the comparator collapses to two
    // f32 compares.
    const float va = fmaf(8.0f, a.x, fmaf(4.0f, a.y, fmaf(2.0f, a.z, a.w)));
    const float vb = fmaf(8.0f, b.x, fmaf(4.0f, b.y, fmaf(2.0f, b.z, b.w)));
    const float gt = (va > vb)  ? 1.0f : 0.0f;
    const float eq = (va == vb) ? 1.0f : 0.0f;

    if (row < n_rows) {
      __builtin_nontemporal_store(gt, out_gt + row);
      __builtin_nontemporal_store(eq, out_eq + row);
    }

    row = next;
    next += stride_rows;
  }
}

extern "C" void kernel_launch(void* const* d_in, const int* in_sizes, int n_in,
                              void* d_out, int out_size, void* d_ws, size_t ws_size,
                              hipStream_t stream) {
  (void)n_in; (void)d_ws; (void)ws_size; (void)out_size;

  const float4* A = (const float4*)d_in[0];
  const float4* B = (const float4*)d_in[1];
  float* out = (float*)d_out;

  const int n_rows = in_sizes[0] / 4;   // (N,4) float32 -> N rows
  if (n_rows <= 0) return;

  const int blocks = 4096;              // 1M threads; N=4.19M -> 4 tiles/thread
  const int stride = blocks * TPB;
  const int iters  = (n_rows + stride - 1) / stride;

  cmp4bit_async_kernel<<<blocks, TPB, 0, stream>>>(
      A, B, out, out + n_rows, n_rows, iters, stride);
}